// SteerableGeometricProductLayer_22462678958647
// MI455X (gfx1250) — compile-verified
//
#include <hip/hip_runtime.h>
#include <hip/hip_bf16.h>
#include <math.h>

typedef __attribute__((ext_vector_type(2))) float v2f;
typedef __attribute__((ext_vector_type(4))) float v4f;
typedef __attribute__((ext_vector_type(8))) float v8f;

static constexpr int kB = 8192;   // batch
static constexpr int kF = 256;    // features

// ---- Cl(3,0) compile-time algebra tables -------------------------------
// Blade order (grade-lex): 1 | e1 e2 e3 | e12 e13 e23 | e123
constexpr int MASKA[8]  = {0, 1, 2, 4, 3, 5, 6, 7};   // blade index -> bitmask
constexpr int IDXOFM[8] = {0, 1, 2, 4, 3, 5, 6, 7};   // bitmask -> blade index
constexpr int GRADE[8]  = {0, 1, 1, 1, 2, 2, 2, 3};

__host__ __device__ constexpr int pcnt(int v) {
  int c = 0; while (v) { c += v & 1; v >>= 1; } return c;
}
__host__ __device__ constexpr float csign(int a, int b) {
  int s = 0; int aa = a >> 1;
  while (aa) { s += pcnt(aa & b); aa >>= 1; }
  return (s & 1) ? -1.0f : 1.0f;
}

// path id for (grade_i, grade_j, grade_k) in np.argwhere (lexicographic) order.
constexpr int PATH_TBL[4][4][4] = {
  // gi = 0
  { { 0,-1,-1,-1}, {-1, 1,-1,-1}, {-1,-1, 2,-1}, {-1,-1,-1, 3} },
  // gi = 1
  { {-1, 4,-1,-1}, { 5,-1, 6,-1}, {-1, 7,-1, 8}, {-1,-1, 9,-1} },
  // gi = 2
  { {-1,-1,10,-1}, {-1,11,-1,12}, {13,-1,14,-1}, {-1,15,-1,-1} },
  // gi = 3
  { {-1,-1,-1,16}, {-1,-1,17,-1}, {-1,18,-1,-1}, {19,-1,-1,-1} },
};

// ---- Kernel 1: repack weights into k-major fragment-friendly layout ----
// Wp[(sg*128 + k/2)*256 + n)*2 + (k&1)] = w_side[n, k, g],  sg = side*4+g
__global__ __launch_bounds__(256)
void repack_w_kernel(const float* __restrict__ wl, const float* __restrict__ wr,
                     float* __restrict__ Wp)
{
  const int o   = blockIdx.x * 256 + threadIdx.x;  // 0 .. 524287 (2 MiB of f32)
  const int par = o & 1;
  const int n   = (o >> 1) & 255;
  const int k2  = (o >> 9) & 127;
  const int sg  = o >> 16;                          // 0..7
  const int side = sg >> 2;
  const int g    = sg & 3;
  const int k    = (k2 << 1) | par;
  const float* src = side ? wr : wl;
  Wp[o] = src[((size_t)n * kF + k) * 4 + g];
}

// ---- Kernel 2: fused WMMA GEMMs + normalize + geometric product --------
// One wave owns a 16(b) x 16(n) tile; 16 f32 accumulators cover
// 8 blades x {left,right}. Epilogue runs entirely in-register.
__global__ __launch_bounds__(256)
void sgp_fused_kernel(const float* __restrict__ x,
                      const float* __restrict__ b_left,
                      const float* __restrict__ a_norm,
                      const float* __restrict__ w_gp,
                      const float* __restrict__ Wp,
                      float* __restrict__ out)
{
  const int lane  = threadIdx.x & 31;
  const int wave  = threadIdx.x >> 5;
  const int btile = blockIdx.x << 4;                       // 512 tiles of 16 rows
  const int ntile = ((blockIdx.y << 3) + wave) << 4;       // 16 tiles of 16 cols
  const int lhi   = lane >> 4;                             // 0/1: K-half & M-half
  const int lm    = lane & 15;                             // M (A) / N (B,C,D)
  const int kk    = lhi << 1;

  const v8f vzero = {0.f,0.f,0.f,0.f,0.f,0.f,0.f,0.f};
  v8f accL[8], accR[8];
#pragma unroll
  for (int i = 0; i < 8; ++i) { accL[i] = vzero; accR[i] = vzero; }

  const float* xrow = x  + (size_t)(btile + lm) * (kF * 8);  // + k*8 per K-step
  const float* wcol = Wp + (size_t)(ntile + lm) * 2;

  for (int k0 = 0; k0 < kF; k0 += 4) {
    const int k = k0 + kk;                     // this lane's K pair start (even)

    // x[b, k, 0..7] and x[b, k+1, 0..7]: 2 x 32 contiguous bytes per lane,
    // feeds the A fragments of all 8 blades at once.
    const v4f* xp = (const v4f*)(xrow + (size_t)k * 8);
    v4f c0 = xp[0];  // x[b,k,  0..3]
    v4f c1 = xp[1];  // x[b,k,  4..7]
    v4f c2 = xp[2];  // x[b,k+1,0..3]
    v4f c3 = xp[3];  // x[b,k+1,4..7]

    v2f aF[8];
#pragma unroll
    for (int i = 0; i < 4; ++i) {
      v2f lo = { c0[i], c2[i] };
      v2f hi = { c1[i], c3[i] };
      aF[i]     = lo;
      aF[i + 4] = hi;
    }

    // B fragments: coalesced float2 (k, k+1) per grade and side.
    const float* wk = wcol + (size_t)(k >> 1) * 512;
    v2f bL[4], bR[4];
#pragma unroll
    for (int g = 0; g < 4; ++g) {
      bL[g] = *(const v2f*)(wk + (size_t)g       * 65536);
      bR[g] = *(const v2f*)(wk + (size_t)(g + 4) * 65536);
    }

#pragma unroll
    for (int i = 0; i < 8; ++i) {
      const int g = GRADE[i];
      accL[i] = __builtin_amdgcn_wmma_f32_16x16x4_f32(
          false, aF[i], false, bL[g], (short)0, accL[i], false, false);
      accR[i] = __builtin_amdgcn_wmma_f32_16x16x4_f32(
          false, aF[i], false, bR[g], (short)0, accR[i], false, false);
    }

    __builtin_prefetch(xrow + (size_t)(k + 4) * 8, 0, 3);
  }

  // ---- epilogue: each lane holds column n for 8 rows b -----------------
  const int n = ntile + lm;

  float sig[4];
#pragma unroll
  for (int g = 0; g < 4; ++g) {
    const float a = a_norm[n * 4 + g];
    sig[g] = 1.0f / (1.0f + __expf(-a));
  }
  float wg[20];
#pragma unroll
  for (int p = 0; p < 20; ++p) wg[p] = w_gp[n * 20 + p];
  const float bias = b_left[n];
  const float RS2  = 0.70710678118654752f;   // 1/sqrt(2)

#pragma unroll
  for (int r = 0; r < 8; ++r) {
    const int b = btile + (lhi << 3) + r;

    const v4f* xq = (const v4f*)(x + ((size_t)b * kF + n) * 8);
    v4f x0 = xq[0], x1 = xq[1];
    float xv[8] = { x0[0], x0[1], x0[2], x0[3], x1[0], x1[1], x1[2], x1[3] };

    float lv[8], rv[8];
#pragma unroll
    for (int i = 0; i < 8; ++i) { lv[i] = accL[i][r]; rv[i] = accR[i][r]; }

    // per-grade smooth norm of the right branch
    float qs[4];
    qs[0] = rv[0]*rv[0];
    qs[1] = rv[1]*rv[1] + rv[2]*rv[2] + rv[3]*rv[3];
    qs[2] = rv[4]*rv[4] + rv[5]*rv[5] + rv[6]*rv[6];
    qs[3] = rv[7]*rv[7];

    float dinv[4];
#pragma unroll
    for (int g = 0; g < 4; ++g) {
      const float nm = sqrtf(sqrtf(qs[g]*qs[g] + 1e-16f));
      dinv[g] = 1.0f / (sig[g] * (nm - 1.0f) + 1.0f + 1e-6f);
    }

    float xr[8];
#pragma unroll
    for (int i = 0; i < 8; ++i) xr[i] = rv[i] * dinv[GRADE[i]];

    // sparse geometric product: 64 Cayley entries, signs & paths fold at
    // compile time via the constexpr tables.
    float gp[8] = {0,0,0,0,0,0,0,0};
#pragma unroll
    for (int i = 0; i < 8; ++i) {
#pragma unroll
      for (int kb = 0; kb < 8; ++kb) {
        const int   mi  = MASKA[i];
        const int   mk  = MASKA[kb];
        const int   j   = IDXOFM[mi ^ mk];
        const float sgn = csign(mi, mk);
        const int   p   = PATH_TBL[GRADE[i]][GRADE[j]][GRADE[kb]];
        gp[j] += sgn * wg[p] * xv[i] * xr[kb];
      }
    }

    float ov[8];
#pragma unroll
    for (int i = 0; i < 8; ++i) {
      const float l = lv[i] + ((i == 0) ? bias : 0.0f);
      ov[i] = (l + gp[i]) * RS2;
    }
    v4f o0 = { ov[0], ov[1], ov[2], ov[3] };
    v4f o1 = { ov[4], ov[5], ov[6], ov[7] };
    v4f* op = (v4f*)(out + ((size_t)b * kF + n) * 8);
    op[0] = o0;
    op[1] = o1;
  }
}

extern "C" void kernel_launch(void* const* d_in, const int* in_sizes, int n_in,
                              void* d_out, int out_size, void* d_ws, size_t ws_size,
                              hipStream_t stream) {
  const float* x       = (const float*)d_in[0];   // [8192,256,8]
  const float* w_left  = (const float*)d_in[1];   // [256,256,4]
  const float* b_left  = (const float*)d_in[2];   // [1,256,1]
  const float* w_right = (const float*)d_in[3];   // [256,256,4]
  const float* a_norm  = (const float*)d_in[4];   // [256,4]
  const float* w_gp    = (const float*)d_in[5];   // [256,20]
  float* out = (float*)d_out;                     // [8192,256,8]
  float* Wp  = (float*)d_ws;                      // 2 MiB repacked weights

  (void)in_sizes; (void)n_in; (void)out_size; (void)ws_size;

  repack_w_kernel<<<dim3(2048), dim3(256), 0, stream>>>(w_left, w_right, Wp);
  sgp_fused_kernel<<<dim3(kB / 16, (kF / 16) / 8), dim3(256), 0, stream>>>(
      x, b_left, a_norm, w_gp, Wp, out);
}